// GraphConv_38113539784821
// MI455X (gfx1250) — compile-verified
//
#include <hip/hip_runtime.h>

typedef __attribute__((ext_vector_type(2))) float v2f;
typedef __attribute__((ext_vector_type(4))) float v4f;
typedef __attribute__((ext_vector_type(8))) float v8f;
typedef int ivec4 __attribute__((vector_size(16)));   // matches builtin's b128 type

#define C_IN   256
#define C_OUT  256
#define NJ     25               // V joints
#define NT     300              // T
#define VT     (NJ * NT)        // 7500
#define M_BLK  64               // output channels per workgroup
#define T_TILE 16               // timesteps per workgroup
#define K_BLK  16               // K staging depth
#define NSTAGE (C_IN / K_BLK)   // 16
#define COLS   (NJ * T_TILE)    // 400
#define WS_STR (K_BLK + 1)      // padded W-tile stride (bank-conflict-free A-frags)
#define WS_FL  (M_BLK * WS_STR) // 1088 floats per W buffer
#define XS_FL  (K_BLK * COLS)   // 6400 floats per X buffer
#define YS_FL  (M_BLK * COLS)   // 25600 floats
#define XS_BASE (2 * WS_FL)     // X buffers start after the two W buffers

#if __has_builtin(__builtin_amdgcn_global_load_async_to_lds_b128) && \
    __has_builtin(__builtin_amdgcn_global_load_async_to_lds_b32)
#define HAS_ASYNC 1
#else
#define HAS_ASYNC 0
#endif

// Cast helpers matching the builtin prototypes:
//   b32 : (global int* src, local int* dst, imm offset, imm cpol)
//   b128: (global ivec4* src, local ivec4* dst, imm offset, imm cpol)
#define G32(p)  ((__attribute__((address_space(1))) int*)(p))
#define L32(p)  ((__attribute__((address_space(3))) int*)(p))
#define G128(p) ((__attribute__((address_space(1))) ivec4*)(p))
#define L128(p) ((__attribute__((address_space(3))) ivec4*)(p))

__device__ __forceinline__ void wait_async0() {
#if HAS_ASYNC
#if __has_builtin(__builtin_amdgcn_s_wait_asynccnt)
  __builtin_amdgcn_s_wait_asynccnt(0);
#else
  asm volatile("s_wait_asynccnt 0x0" ::: "memory");
#endif
#endif
}

// Issue one K_BLK stage of W and X tiles into LDS (async DMA when available).
__device__ __forceinline__ void stage_load(const float* __restrict__ W,
                                           const float* __restrict__ xN,
                                           float* Ws, float* Xs,
                                           int mb, int t0, int k0, int tid,
                                           bool tail) {
#pragma unroll
  for (int i = 0; i < (M_BLK * K_BLK) / 256; ++i) {   // 4 per thread
    int e = tid + i * 256;
    int m = e >> 4, kk = e & 15;
    const float* src = &W[(size_t)(mb + m) * C_IN + (k0 + kk)];
    float* dst = &Ws[m * WS_STR + kk];
#if HAS_ASYNC
    __builtin_amdgcn_global_load_async_to_lds_b32(G32(src), L32(dst), 0, 0);
#else
    *dst = *src;
#endif
  }
#pragma unroll
  for (int i = 0; i < 7; ++i) {                        // 1600 float4 / 256 thr
    int e4 = tid + i * 256;
    if (e4 < XS_FL / 4) {
      int k = e4 / 100, r = e4 - k * 100;              // 100 float4 per k-row
      int v = r >> 2, ttg = (r & 3) * 4;               // col = v*16 + ttg..+3
      if (!tail || (t0 + ttg + 3 < NT)) {              // OOB groups stay zero
        const float* src = &xN[(size_t)(k0 + k) * VT + v * NT + t0 + ttg];
        float* dst = &Xs[e4 * 4];
#if HAS_ASYNC
        __builtin_amdgcn_global_load_async_to_lds_b128(G128(src), L128(dst), 0, 0);
#else
        *(v4f*)dst = *(const v4f*)src;
#endif
      }
    }
  }
}

// Fused: per-batch channel GEMM Y = W*X via V_WMMA_F32_16X16X4_F32,
// then joint mix out[.,v,t] = b + sum_w A[w,v] * Y[.,w,t].
__global__ __launch_bounds__(256)
void gconv_wmma_f32(const float* __restrict__ x, const float* __restrict__ A,
                    const float* __restrict__ W, const float* __restrict__ bias,
                    float* __restrict__ out) {
  __shared__ float sA[NJ * NJ];
  __shared__ float sR[YS_FL];                 // Ys region; staging aliased inside
  float* Ys = sR;

  const int tid = threadIdx.x;
  const int n   = blockIdx.x;
  const int mb  = blockIdx.y * M_BLK;
  const int t0  = blockIdx.z * T_TILE;
  const bool tail = (t0 + T_TILE > NT);

  const int lane  = tid & 31;
  const int wave  = __builtin_amdgcn_readfirstlane(tid >> 5);
  const int mp    = wave >> 2;                       // 0..1 -> two 16-row tiles
  const int q     = wave & 3;                        // joint quarter
  const int v_beg = (q == 0) ? 0 : (7 + (q - 1) * 6);
  const int vcnt  = (q == 0) ? 7 : 6;                // 7+6+6+6 = 25
  const int nloc  = lane & 15;
  const int khalf = (lane >> 4) << 1;
  const int mrow0 = mp * 32 + nloc;
  const int mrow1 = mp * 32 + 16 + nloc;

  if (tail)                                          // pre-zero both X buffers
    for (int e = tid; e < 2 * XS_FL; e += 256) sR[XS_BASE + e] = 0.0f;
  for (int e = tid; e < NJ * NJ; e += 256) sA[e] = A[e];
  __syncthreads();

  const float* xN = x + (size_t)n * C_IN * VT;

  stage_load(W, xN, sR, sR + XS_BASE, mb, t0, 0, tid, tail);
  wait_async0();
  __syncthreads();

  v8f acc0[7] = {};
  v8f acc1[7] = {};

  for (int s = 0; s < NSTAGE; ++s) {
    const int cur = s & 1;
    if (s + 1 < NSTAGE) {                            // prefetch next stage (async)
      const int nb = (s + 1) & 1;
      stage_load(W, xN, sR + nb * WS_FL, sR + XS_BASE + nb * XS_FL,
                 mb, t0, (s + 1) * K_BLK, tid, tail);
    }

    const float* ws = sR + cur * WS_FL;
    const float* xs = sR + XS_BASE + cur * XS_FL;
#pragma unroll
    for (int kk = 0; kk < K_BLK; kk += 4) {
      const int kk2 = kk + khalf;
      v2f af0, af1;
      af0.x = ws[mrow0 * WS_STR + kk2];
      af0.y = ws[mrow0 * WS_STR + kk2 + 1];
      af1.x = ws[mrow1 * WS_STR + kk2];
      af1.y = ws[mrow1 * WS_STR + kk2 + 1];
#pragma unroll
      for (int i = 0; i < 7; ++i) {
        if (i < vcnt) {                              // wave-uniform guard
          const int col = (v_beg + i) * 16 + nloc;
          v2f bf;
          bf.x = xs[kk2 * COLS + col];
          bf.y = xs[(kk2 + 1) * COLS + col];
          acc0[i] = __builtin_amdgcn_wmma_f32_16x16x4_f32(
              false, af0, false, bf, (short)0, acc0[i], false, false);
          acc1[i] = __builtin_amdgcn_wmma_f32_16x16x4_f32(
              false, af1, false, bf, (short)0, acc1[i], false, false);
        }
      }
    }
    wait_async0();
    __syncthreads();
  }

  // spill Y tiles to LDS (C/D layout: VGPR e -> M = e + 8*(lane>=16), N = lane&15)
  {
    const int mh = (lane >> 4) * 8;
#pragma unroll
    for (int i = 0; i < 7; ++i) {
      if (i < vcnt) {
        const int colb = (v_beg + i) * 16 + nloc;
#pragma unroll
        for (int e = 0; e < 8; ++e) {
          Ys[(mp * 32 + mh + e) * COLS + colb]      = acc0[i][e];
          Ys[(mp * 32 + 16 + mh + e) * COLS + colb] = acc1[i][e];
        }
      }
    }
  }
  __syncthreads();

  // joint mix: one thread per (m, t-quad); Y row cached in registers (25 x b128)
  {
    const int m   = tid >> 2;
    const int ttg = (tid & 3) * 4;
    v4f Y4[NJ];
#pragma unroll
    for (int w = 0; w < NJ; ++w)
      Y4[w] = *(const v4f*)&Ys[m * COLS + w * 16 + ttg];
    const float bb = bias[mb + m];
    const bool sok = !tail || (t0 + ttg + 3 < NT);
    float* obase = out + ((size_t)n * C_OUT + mb + m) * VT + t0 + ttg;
#pragma unroll 1
    for (int v = 0; v < NJ; ++v) {
      v4f s4 = {bb, bb, bb, bb};
#pragma unroll
      for (int w = 0; w < NJ; ++w)
        s4 += sA[w * NJ + v] * Y4[w];
      if (sok) *(v4f*)(obase + v * NT) = s4;
    }
  }
}

extern "C" void kernel_launch(void* const* d_in, const int* in_sizes, int n_in,
                              void* d_out, int out_size, void* d_ws, size_t ws_size,
                              hipStream_t stream) {
  (void)in_sizes; (void)n_in; (void)out_size; (void)d_ws; (void)ws_size;
  const float* x = (const float*)d_in[0];
  const float* A = (const float*)d_in[1];
  const float* W = (const float*)d_in[2];
  const float* b = (const float*)d_in[3];
  float* out = (float*)d_out;

  dim3 grid(32, C_OUT / M_BLK, (NT + T_TILE - 1) / T_TILE);  // 32 x 4 x 19
  gconv_wmma_f32<<<grid, 256, 0, stream>>>(x, A, W, b, out);
}